// ModelNew_3556232922034
// MI455X (gfx1250) — compile-verified
//
#include <hip/hip_runtime.h>
#include <hip/hip_bf16.h>
#include <math.h>

typedef float v2f __attribute__((ext_vector_type(2)));
typedef float v8f __attribute__((ext_vector_type(8)));

#define C_IN   64
#define C_OUT  64
#define H_IN   64
#define W_IN   64
#define H_OUT  128
#define W_OUT  128
#define KS     4
#define H_ACT  34      // rows with any conv contribution (2h-kh <= 63, kh<=3 -> h<=33)
#define LDS_PAD 66     // ci dimension padded 64 -> 66 to kill LDS bank conflicts

__device__ __forceinline__ float fused_act(float y) {
    // y already includes bias + 0.5
    y = fminf(y, 0.0f);
    float g = 0.5f * y * (1.0f + tanhf(0.7978845608f * (y + 0.044715f * y * y * y)));
    return g * 2.0f;
}

// ---------------------------------------------------------------------------
// Kernel 1: fill entire output with the per-channel constant f(bias[c]+0.5).
// Pure bandwidth: one float4 (global_store_b128) per thread.
// ---------------------------------------------------------------------------
__global__ __launch_bounds__(256) void ctgelu_fill_kernel(
        const float* __restrict__ bias, float* __restrict__ out) {
    size_t idx = (size_t)blockIdx.x * 256 + threadIdx.x;     // float4 index
    int c = (int)((idx >> 12) & 63);                         // (idx*4 / 16384) % 64
    float v = fused_act(bias[c] + 0.5f);
    float4 o; o.x = v; o.y = v; o.z = v; o.w = v;
    ((float4*)out)[idx] = o;
}

// ---------------------------------------------------------------------------
// Kernel 2: compute the active corner h in [0,34), w in [0,48) via fp32 WMMA.
// One block per (h, n): 128 threads = 4 waves, wave w owns c_out block 16w.
// LDS holds x[n, :, 2h-kh, :] transposed to xs[kh][wi][ci] (ci padded to 66).
// GEMM: D[16 c_out x 16 w] += A[16 c_out x 4 ci] * B[4 ci x 16 w], 16 K-chunks
// x 16 taps, using V_WMMA_F32_16X16X4_F32 (full fp32 precision).
// For the t==2 tile (w in [32,48)) the kw==0 taps have B identically zero
// (wi = 64+2n >= 64 for every lane), so they are elided at compile time.
// ---------------------------------------------------------------------------
__global__ __launch_bounds__(128) void ctgelu_wmma_kernel(
        const float* __restrict__ x, const float* __restrict__ wgt,
        const float* __restrict__ bias, float* __restrict__ out) {
    extern __shared__ float xs[];            // [4][64][LDS_PAD] floats = 67584 B

    const int h    = blockIdx.x;             // 0..33
    const int n    = blockIdx.y;             // 0..63
    const int tid  = threadIdx.x;            // 0..127
    const int lane = tid & 31;
    const int wv   = tid >> 5;               // wave id 0..3
    const int ln   = lane & 15;              // N / M-row position
    const int hi   = lane >> 4;              // lane half: selects K {0,1} vs {2,3}
    const int c0   = hi * 2;                 // K base within a 4-chunk

    // ---- stage 4 input rows (all 64 ci) into LDS, transposed + zero-padded ----
    for (int r = tid; r < 4 * C_IN; r += 128) {
        const int kh = r >> 6;
        const int ci = r & 63;
        const int iy = 2 * h - kh;
        float* dst = xs + (size_t)(kh * W_IN) * LDS_PAD + ci;   // + wi*LDS_PAD
        if (iy >= 0 && iy < H_IN) {
            const float4* src =
                (const float4*)(x + (((size_t)n * C_IN + ci) * H_IN + iy) * W_IN);
            #pragma unroll
            for (int q = 0; q < W_IN / 4; ++q) {
                float4 v = src[q];
                dst[(4 * q + 0) * LDS_PAD] = v.x;
                dst[(4 * q + 1) * LDS_PAD] = v.y;
                dst[(4 * q + 2) * LDS_PAD] = v.z;
                dst[(4 * q + 3) * LDS_PAD] = v.w;
            }
        } else {
            #pragma unroll
            for (int wi = 0; wi < W_IN; ++wi) dst[wi * LDS_PAD] = 0.0f;
        }
    }
    __syncthreads();

    // ---- per-wave GEMM over 3 w-tiles x 16 taps x 16 K-chunks ----
    const int m0 = wv * 16;                  // c_out block base
    const v2f vz = {};
    v8f acc[3] = {};

    // weight base for this lane's A row (M = m0 + ln)
    const float* wrow = wgt + (size_t)(m0 + ln) * (C_IN * KS * KS);
    __builtin_prefetch(wrow, 0, 0);

    #pragma unroll
    for (int kh = 0; kh < KS; ++kh) {
        #pragma unroll
        for (int kw = 0; kw < KS; ++kw) {
            // A fragments for this tap: a[kc] = (W[m, c0+4kc], W[m, c0+4kc+1])
            const float* wp = wrow + kh * KS + kw;
            v2f a[16];
            #pragma unroll
            for (int kc = 0; kc < 16; ++kc) {
                const int ci = 4 * kc + c0;
                a[kc].x = wp[(size_t)ci * (KS * KS)];
                a[kc].y = wp[(size_t)(ci + 1) * (KS * KS)];
            }
            #pragma unroll
            for (int t = 0; t < 3; ++t) {
                // Exact no-op elision: t==2, kw==0 -> every lane out of range.
                if (t == 2 && kw == 0) continue;     // compile-time constant fold
                const int wo = t * 16 + ln;          // output column this lane owns
                const int wi = 2 * wo - kw;          // input column
                const bool valid = (wi >= 0) && (wi < W_IN);
                const int wiu = valid ? wi : 0;      // clamped (select, no branch)
                const float* bp = xs + (size_t)(kh * W_IN + wiu) * LDS_PAD + c0;
                #pragma unroll
                for (int kc = 0; kc < 16; ++kc) {
                    v2f b = *(const v2f*)(bp + 4 * kc);   // ds_load_b64: (ci, ci+1)
                    b = valid ? b : vz;                   // edge mask via cndmask
                    acc[t] = __builtin_amdgcn_wmma_f32_16x16x4_f32(
                        false, a[kc], false, b, (short)0, acc[t], false, false);
                }
            }
        }
    }

    // ---- epilogue: bias + 0.5, clamp-min-0, tanh-GELU, x2, store ----
    float bv[8];
    #pragma unroll
    for (int v = 0; v < 8; ++v) bv[v] = bias[m0 + hi * 8 + v];

    #pragma unroll
    for (int t = 0; t < 3; ++t) {
        #pragma unroll
        for (int v = 0; v < 8; ++v) {
            const int cout = m0 + hi * 8 + v;        // D-matrix M for this VGPR/half
            const int wcol = t * 16 + ln;            // D-matrix N
            float r = fused_act(acc[t][v] + bv[v] + 0.5f);
            out[(((size_t)n * C_OUT + cout) * H_OUT + h) * W_OUT + wcol] = r;
        }
    }
}

extern "C" void kernel_launch(void* const* d_in, const int* in_sizes, int n_in,
                              void* d_out, int out_size, void* d_ws, size_t ws_size,
                              hipStream_t stream) {
    const float* x    = (const float*)d_in[0];   // [64,64,64,64]
    const float* wgt  = (const float*)d_in[1];   // [64,64,4,4]
    const float* bias = (const float*)d_in[2];   // [64]
    float* out        = (float*)d_out;           // [64,64,128,128]

    // 1) constant fill of the whole output (bandwidth-bound floor)
    const size_t n4 = (size_t)64 * 64 * 128 * 128 / 4;   // 16,777,216 float4
    ctgelu_fill_kernel<<<dim3((unsigned)(n4 / 256)), dim3(256), 0, stream>>>(bias, out);

    // 2) WMMA compute of the active corner (h<34, w<48); overwrites fill there
    const size_t lds_bytes = (size_t)4 * W_IN * LDS_PAD * sizeof(float);  // 67584
    ctgelu_wmma_kernel<<<dim3(H_ACT, 64), dim3(128), lds_bytes, stream>>>(
        x, wgt, bias, out);
}